// GCN_37108517437514
// MI455X (gfx1250) — compile-verified
//
#include <hip/hip_runtime.h>

typedef float v2f __attribute__((ext_vector_type(2)));
typedef float v8f __attribute__((ext_vector_type(8)));

#define HD 128         // hidden dim
#define NG 1000        // graphs
#define OD 10          // output dim

// ---------------- small utility kernels ----------------

__global__ void fill_kernel(float* __restrict__ p, float v, int n) {
    int i = blockIdx.x * blockDim.x + threadIdx.x;
    if (i < n) p[i] = v;
}

// deg[dst] += 1 for every edge (deg pre-initialized to 1.0 for the self loop)
__global__ void deg_edges_kernel(const int* __restrict__ ei, float* __restrict__ deg, int E) {
    int e = blockIdx.x * blockDim.x + threadIdx.x;
    if (e < E) {
        int dst = ei[E + e];
        unsafeAtomicAdd(deg + dst, 1.0f);
    }
}

// in-place deg -> dinv = rsqrt(deg)   (deg >= 1 always, so reference's where/max are no-ops)
__global__ void rsqrt_kernel(float* __restrict__ p, int n) {
    int i = blockIdx.x * blockDim.x + threadIdx.x;
    if (i < n) p[i] = rsqrtf(p[i]);
}

// ---------------- FP32 WMMA GEMM with fused GCN-norm epilogue ----------------
// Computes HW = A[N,K] @ W[K,128]; writes HS = HW*dinv[row] and OUT = HW*dinv[row]^2
// (OUT is the self-loop-initialized scatter accumulator).
//
// W is staged in LDS in k-pair-major layout: element (k,c) lives at float offset
// ((k>>1)*HD + c)*2 + (k&1), so every WMMA B-fragment (two consecutive k's, same
// column) is one aligned 8-byte ds_load_b64 straight into a register pair.
template <int K>
__global__ __launch_bounds__(128)
void gemm_gcn_kernel(const float* __restrict__ A, const float* __restrict__ W,
                     const float* __restrict__ dinv,
                     float* __restrict__ HS, float* __restrict__ OUT, int N) {
    __shared__ float sW[K * HD];
    for (int i = threadIdx.x; i < K * HD / 4; i += blockDim.x) {
        float4 v = ((const float4*)W)[i];
        int j = i * 4;
        int k = j / HD;            // W row (k index)
        int c = j % HD;            // W col
        float* dp = sW + ((size_t)(k >> 1) * HD + c) * 2 + (k & 1);
        dp[0] = v.x;
        dp[2] = v.y;
        dp[4] = v.z;
        dp[6] = v.w;
    }
    __syncthreads();

    const int lane  = threadIdx.x & 31;
    const int wave  = threadIdx.x >> 5;
    const int rowBase = (blockIdx.x * 4 + wave) * 16;
    const int m     = lane & 15;     // row-in-tile for A, col-in-16 for B/C
    const int khalf = lane >> 4;     // K half selector

    int row  = rowBase + m;
    int rowc = row < N ? row : N - 1;              // clamp so loads stay in-bounds

    v8f acc[8] = {};                               // 16x128 tile = 8 accumulators

    const float* arow  = A + (size_t)rowc * K + 2 * khalf;
    const float* bbase = sW + ((size_t)khalf * HD + m) * 2;
    for (int kb = 0; kb < K; kb += 4) {
        // A fragment: reg r holds A[row][kb + 2*khalf + r]  (one 8B load)
        v2f af = *(const v2f*)(arow + kb);
        const float* bp = bbase + (size_t)(kb >> 1) * HD * 2;
#pragma unroll
        for (int t = 0; t < 8; ++t) {
            // B fragment: reg r holds W[kb + 2*khalf + r][t*16 + m]  (one ds_load_b64)
            v2f bf = *(const v2f*)(bp + t * 32);
            acc[t] = __builtin_amdgcn_wmma_f32_16x16x4_f32(
                false, af, false, bf, (short)0, acc[t], false, false);
        }
    }

    // C/D layout: reg r -> output row (r + 8*khalf), col (t*16 + m)
    float sr[8];
#pragma unroll
    for (int r = 0; r < 8; ++r) {
        int orow = rowBase + r + 8 * khalf;
        sr[r] = (orow < N) ? dinv[orow] : 0.0f;
    }
#pragma unroll
    for (int t = 0; t < 8; ++t) {
#pragma unroll
        for (int r = 0; r < 8; ++r) {
            int orow = rowBase + r + 8 * khalf;
            if (orow < N) {
                float v = acc[t][r];
                float s = sr[r];
                size_t idx = (size_t)orow * HD + t * 16 + m;
                HS[idx]  = v * s;        // h@W scaled by dinv[src] (edge message source)
                OUT[idx] = v * s * s;    // self-loop contribution
            }
        }
    }
}

// ---------------- edge scatter: OUT[dst] += HS[src] * dinv[dst] ----------------
// one wave (32 lanes) per edge; 4 floats per lane via float4 gather + 4 f32 atomics
__global__ void edge_scatter_kernel(const int* __restrict__ ei,
                                    const float* __restrict__ HS,
                                    const float* __restrict__ dinv,
                                    float* __restrict__ OUT, int E) {
    int tid = blockIdx.x * blockDim.x + threadIdx.x;
    int e = tid >> 5;
    if (e >= E) return;
    int lane = tid & 31;
    int src = ei[e];
    int dst = ei[E + e];
    float w = dinv[dst];
    const float4 v = *(const float4*)(HS + (size_t)src * HD + lane * 4);
    float* o = OUT + (size_t)dst * HD + lane * 4;
    unsafeAtomicAdd(o + 0, v.x * w);
    unsafeAtomicAdd(o + 1, v.y * w);
    unsafeAtomicAdd(o + 2, v.z * w);
    unsafeAtomicAdd(o + 3, v.w * w);
}

// ---------------- bias + relu ----------------
__global__ void bias_relu_kernel(const float* __restrict__ in, const float* __restrict__ b,
                                 float* __restrict__ out, int total) {
    int i = blockIdx.x * blockDim.x + threadIdx.x;
    if (i < total) {
        float v = in[i] + b[i & (HD - 1)];
        out[i] = v > 0.0f ? v : 0.0f;
    }
}

// ---------------- global mean pool (accumulation) ----------------
// one wave per node: psum[batch[n]] += h[n]; cnt[batch[n]] += 1
__global__ void pool_scatter_kernel(const int* __restrict__ batch,
                                    const float* __restrict__ h,
                                    float* __restrict__ psum, float* __restrict__ cnt, int N) {
    int tid = blockIdx.x * blockDim.x + threadIdx.x;
    int n = tid >> 5;
    if (n >= N) return;
    int lane = tid & 31;
    int g = batch[n];
    const float4 v = *(const float4*)(h + (size_t)n * HD + lane * 4);
    float* o = psum + (size_t)g * HD + lane * 4;
    unsafeAtomicAdd(o + 0, v.x);
    unsafeAtomicAdd(o + 1, v.y);
    unsafeAtomicAdd(o + 2, v.z);
    unsafeAtomicAdd(o + 3, v.w);
    if (lane == 0) unsafeAtomicAdd(cnt + g, 1.0f);
}

// ---------------- final linear: out[g] = (psum[g]/max(cnt,1)) @ Wl + bl ----------------
__global__ void final_linear_kernel(const float* __restrict__ psum, const float* __restrict__ cnt,
                                    const float* __restrict__ Wl, const float* __restrict__ bl,
                                    float* __restrict__ out) {
    int g = blockIdx.x;
    int j = threadIdx.x;
    if (j < OD) {
        float inv = 1.0f / fmaxf(cnt[g], 1.0f);
        float acc = bl[j];
        const float* p = psum + (size_t)g * HD;
        for (int k = 0; k < HD; ++k) acc += p[k] * inv * Wl[k * OD + j];
        out[g * OD + j] = acc;
    }
}

// ---------------- launcher ----------------
extern "C" void kernel_launch(void* const* d_in, const int* in_sizes, int n_in,
                              void* d_out, int out_size, void* d_ws, size_t ws_size,
                              hipStream_t stream) {
    const float* x   = (const float*)d_in[0];   // [N,64]
    const int*   ei  = (const int*)d_in[1];     // [2,E]
    const int*   bat = (const int*)d_in[2];     // [N]
    const float* W1  = (const float*)d_in[3];
    const float* b1  = (const float*)d_in[4];
    const float* W2  = (const float*)d_in[5];
    const float* b2  = (const float*)d_in[6];
    const float* W3  = (const float*)d_in[7];
    const float* b3  = (const float*)d_in[8];
    const float* Wl  = (const float*)d_in[9];
    const float* bl  = (const float*)d_in[10];
    float* out = (float*)d_out;

    const int N = in_sizes[0] / 64;
    const int E = in_sizes[1] / 2;

    // workspace carve-up (floats)
    float* ws   = (float*)d_ws;
    float* dinv = ws;                          // N          (deg, then rsqrt'd in place)
    float* HS   = dinv + N;                    // N*128
    float* OUT  = HS + (size_t)N * HD;         // N*128
    float* Abuf = OUT + (size_t)N * HD;        // N*128
    float* psum = Abuf + (size_t)N * HD;       // NG*128
    float* cnt  = psum + (size_t)NG * HD;      // NG   (contiguous after psum)

    const int TB = 256;
    // degree / normalization
    fill_kernel<<<(N + TB - 1) / TB, TB, 0, stream>>>(dinv, 1.0f, N);
    deg_edges_kernel<<<(E + TB - 1) / TB, TB, 0, stream>>>(ei, dinv, E);
    rsqrt_kernel<<<(N + TB - 1) / TB, TB, 0, stream>>>(dinv, N);

    const int gemmGrid = (N + 63) / 64;
    const int edgeGrid = (int)(((size_t)E * 32 + TB - 1) / TB);
    const int nodeGrid = (int)(((size_t)N * HD + TB - 1) / TB);
    const int poolGrid = (int)(((size_t)N * 32 + TB - 1) / TB);

    // layer 1 (K = 64, input x)
    gemm_gcn_kernel<64><<<gemmGrid, 128, 0, stream>>>(x, W1, dinv, HS, OUT, N);
    edge_scatter_kernel<<<edgeGrid, TB, 0, stream>>>(ei, HS, dinv, OUT, E);
    bias_relu_kernel<<<nodeGrid, TB, 0, stream>>>(OUT, b1, Abuf, N * HD);

    // layer 2 (K = 128)
    gemm_gcn_kernel<128><<<gemmGrid, 128, 0, stream>>>(Abuf, W2, dinv, HS, OUT, N);
    edge_scatter_kernel<<<edgeGrid, TB, 0, stream>>>(ei, HS, dinv, OUT, E);
    bias_relu_kernel<<<nodeGrid, TB, 0, stream>>>(OUT, b2, Abuf, N * HD);

    // layer 3 (K = 128)
    gemm_gcn_kernel<128><<<gemmGrid, 128, 0, stream>>>(Abuf, W3, dinv, HS, OUT, N);
    edge_scatter_kernel<<<edgeGrid, TB, 0, stream>>>(ei, HS, dinv, OUT, E);
    bias_relu_kernel<<<nodeGrid, TB, 0, stream>>>(OUT, b3, Abuf, N * HD);

    // global mean pool + classifier head
    fill_kernel<<<(NG * HD + NG + TB - 1) / TB, TB, 0, stream>>>(psum, 0.0f, NG * HD + NG);
    pool_scatter_kernel<<<poolGrid, TB, 0, stream>>>(bat, Abuf, psum, cnt, N);
    final_linear_kernel<<<NG, 32, 0, stream>>>(psum, cnt, Wl, bl, out);
}